// DualGATNet_75445395522171
// MI455X (gfx1250) — compile-verified
//
#include <hip/hip_runtime.h>
#include <hip/hip_bf16.h>

typedef __attribute__((ext_vector_type(2))) float v2f;
typedef __attribute__((ext_vector_type(8))) float v8f;

#define D 64
#define XS 65            // padded LDS row stride (floats) to avoid bank conflicts
#define NEG_SLOPE 0.2f

// ---------------------------------------------------------------------------
// GEMM: h = x @ W  (N x 64 @ 64 x 64), plus alpha_s = h@a_src, alpha_d = h@a_dst
// Block = 256 threads (8 waves), each wave computes a 16x64 tile of h via
// V_WMMA_F32_16X16X4_F32 (16 K-steps x 4 N-tiles = 64 WMMA per wave).
// ---------------------------------------------------------------------------
__global__ __launch_bounds__(256) void gat_gemm(
    const float* __restrict__ x, const float* __restrict__ W,
    const float* __restrict__ asrc, const float* __restrict__ adst,
    float* __restrict__ h, float* __restrict__ alps, float* __restrict__ alpd,
    int N)
{
    __shared__ float sW[D * D];        // 16 KB
    __shared__ float sX[128 * XS];     // 33 KB (x tile, later reused as h tile)
    __shared__ float sV[2 * D];        // asrc | adst

    const int tid  = threadIdx.x;
    const int row0 = blockIdx.x * 128;

    for (int i = tid; i < D * D; i += 256) sW[i] = W[i];
    if (tid < 2 * D) sV[tid] = (tid < D) ? asrc[tid] : adst[tid - D];
    for (int i = tid; i < 128 * D; i += 256) {
        int r = i >> 6, c = i & 63;
        int row = row0 + r;
        sX[r * XS + c] = (row < N) ? x[(size_t)row * D + c] : 0.0f;
    }
    __syncthreads();

    const int wave  = tid >> 5;
    const int lane  = tid & 31;
    const int half  = lane >> 4;   // 0: K pair {0,1} / rows M, 1: K pair {2,3} / rows M+8
    const int m     = lane & 15;
    const int rbase = wave * 16;   // local row base of this wave's 16x64 tile

    v8f acc0 = {}, acc1 = {}, acc2 = {}, acc3 = {};

    #pragma unroll
    for (int k0 = 0; k0 < D; k0 += 4) {
        const int ka = k0 + 2 * half;
        v2f a;
        a.x = sX[(rbase + m) * XS + ka];
        a.y = sX[(rbase + m) * XS + ka + 1];
        v2f b0, b1, b2, b3;
        b0.x = sW[ka * D +  0 + m]; b0.y = sW[(ka + 1) * D +  0 + m];
        b1.x = sW[ka * D + 16 + m]; b1.y = sW[(ka + 1) * D + 16 + m];
        b2.x = sW[ka * D + 32 + m]; b2.y = sW[(ka + 1) * D + 32 + m];
        b3.x = sW[ka * D + 48 + m]; b3.y = sW[(ka + 1) * D + 48 + m];
        acc0 = __builtin_amdgcn_wmma_f32_16x16x4_f32(false, a, false, b0, (short)0, acc0, false, false);
        acc1 = __builtin_amdgcn_wmma_f32_16x16x4_f32(false, a, false, b1, (short)0, acc1, false, false);
        acc2 = __builtin_amdgcn_wmma_f32_16x16x4_f32(false, a, false, b2, (short)0, acc2, false, false);
        acc3 = __builtin_amdgcn_wmma_f32_16x16x4_f32(false, a, false, b3, (short)0, acc3, false, false);
    }

    // C/D layout: vgpr r holds row (r + 8*half), column (16*t + m) at this lane.
    // Each wave writes only its own 16 LDS rows, so no barrier needed before
    // overwriting its slice of sX with h.
    #pragma unroll
    for (int r = 0; r < 8; ++r) {
        const int rr  = rbase + r + 8 * half;
        const int row = row0 + rr;
        float v0 = acc0[r], v1 = acc1[r], v2 = acc2[r], v3 = acc3[r];
        sX[rr * XS +  0 + m] = v0;
        sX[rr * XS + 16 + m] = v1;
        sX[rr * XS + 32 + m] = v2;
        sX[rr * XS + 48 + m] = v3;
        if (row < N) {
            float* hp = h + (size_t)row * D;
            hp[ 0 + m] = v0; hp[16 + m] = v1; hp[32 + m] = v2; hp[48 + m] = v3;
        }
    }
    __syncthreads();

    if (tid < 128) {
        const int row = row0 + tid;
        if (row < N) {
            float ss = 0.f, dd = 0.f;
            #pragma unroll
            for (int c = 0; c < D; ++c) {
                float v = sX[tid * XS + c];
                ss += v * sV[c];
                dd += v * sV[D + c];
            }
            alps[row] = ss;
            alpd[row] = dd;
        }
    }
}

// ---------------------------------------------------------------------------
// Segment softmax + aggregation over edges (self-loops appended as idx >= E).
// Float max via monotonic ordered-uint encoding + atomicMax.
// ---------------------------------------------------------------------------
__device__ __forceinline__ unsigned f2ord(float f) {
    unsigned u = __float_as_uint(f);
    return (u & 0x80000000u) ? ~u : (u | 0x80000000u);
}
__device__ __forceinline__ float ord2f(unsigned e) {
    return (e & 0x80000000u) ? __uint_as_float(e & 0x7FFFFFFFu) : __uint_as_float(~e);
}

__global__ void init_seg(float* __restrict__ agg, unsigned* __restrict__ mmax,
                         float* __restrict__ ssum, int N)
{
    int i = blockIdx.x * 256 + threadIdx.x;
    if (i < N * D) agg[i] = 0.0f;
    if (i < N) { mmax[i] = 0u; ssum[i] = 0.0f; }   // ord(x) > 0 for all finite x
}

__global__ void edge_max_k(const int* __restrict__ src, const int* __restrict__ dst,
                           const float* __restrict__ as, const float* __restrict__ ad,
                           unsigned* __restrict__ mmax, int E, int N)
{
    int i = blockIdx.x * 256 + threadIdx.x;
    if (i >= E + N) return;
    int si = (i < E) ? src[i] : (i - E);
    int di = (i < E) ? dst[i] : (i - E);
    float e = as[si] + ad[di];
    e = (e > 0.f) ? e : NEG_SLOPE * e;
    atomicMax(&mmax[di], f2ord(e));
}

__global__ void edge_sum_k(const int* __restrict__ src, const int* __restrict__ dst,
                           const float* __restrict__ as, const float* __restrict__ ad,
                           const unsigned* __restrict__ mmax, float* __restrict__ ssum,
                           int E, int N)
{
    int i = blockIdx.x * 256 + threadIdx.x;
    if (i >= E + N) return;
    int si = (i < E) ? src[i] : (i - E);
    int di = (i < E) ? dst[i] : (i - E);
    float e = as[si] + ad[di];
    e = (e > 0.f) ? e : NEG_SLOPE * e;
    atomicAdd(&ssum[di], expf(e - ord2f(mmax[di])));
}

// 16 threads per edge, float4 gathers of h[src], f32 atomic adds into agg[dst].
__global__ void edge_agg_k(const int* __restrict__ src, const int* __restrict__ dst,
                           const float* __restrict__ as, const float* __restrict__ ad,
                           const unsigned* __restrict__ mmax, const float* __restrict__ ssum,
                           const float* __restrict__ h, float* __restrict__ agg,
                           int E, int N)
{
    int gid = blockIdx.x * 256 + threadIdx.x;
    int i = gid >> 4;
    int q = gid & 15;
    if (i >= E + N) return;
    int si = (i < E) ? src[i] : (i - E);
    int di = (i < E) ? dst[i] : (i - E);
    float e = as[si] + ad[di];
    e = (e > 0.f) ? e : NEG_SLOPE * e;
    float alpha = expf(e - ord2f(mmax[di])) / ssum[di];
    float4 hv = *(const float4*)(h + (size_t)si * D + q * 4);
    float* o = agg + (size_t)di * D + q * 4;
    atomicAdd(o + 0, hv.x * alpha);
    atomicAdd(o + 1, hv.y * alpha);
    atomicAdd(o + 2, hv.z * alpha);
    atomicAdd(o + 3, hv.w * alpha);
}

__global__ void combine2_k(float* __restrict__ x2out, const float* __restrict__ bias, int N)
{
    int i = blockIdx.x * 256 + threadIdx.x;
    if (i < N * D) x2out[i] += bias[i & 63];
}

__global__ void combine1_k(float* __restrict__ x1out, const float* __restrict__ bias,
                           const float* __restrict__ x2out, const int* __restrict__ grp,
                           int N1)
{
    int i = blockIdx.x * 256 + threadIdx.x;
    if (i < N1 * D) {
        int r = i >> 6, c = i & 63;
        x1out[i] += bias[c] + x2out[(size_t)grp[r] * D + c];
    }
}

// ---------------------------------------------------------------------------
extern "C" void kernel_launch(void* const* d_in, const int* in_sizes, int n_in,
                              void* d_out, int out_size, void* d_ws, size_t ws_size,
                              hipStream_t stream)
{
    const float* X1  = (const float*)d_in[0];
    const int*   EI1 = (const int*)  d_in[1];
    const float* X2  = (const float*)d_in[2];
    const int*   EI2 = (const int*)  d_in[3];
    const int*   GRP = (const int*)  d_in[4];
    const float* W1  = (const float*)d_in[5];
    const float* AS1 = (const float*)d_in[6];
    const float* AD1 = (const float*)d_in[7];
    const float* B1  = (const float*)d_in[8];
    const float* W2  = (const float*)d_in[9];
    const float* AS2 = (const float*)d_in[10];
    const float* AD2 = (const float*)d_in[11];
    const float* B2  = (const float*)d_in[12];

    const int N1 = in_sizes[0] / D;
    const int E1 = in_sizes[1] / 2;
    const int N2 = in_sizes[2] / D;
    const int E2 = in_sizes[3] / 2;
    const int L  = in_sizes[5] / (D * D);

    const int* src1 = EI1;
    const int* dst1 = EI1 + E1;
    const int* src2 = EI2;
    const int* dst2 = EI2 + E2;

    // Workspace carve-up (256B aligned)
    char* p = (char*)d_ws;
    auto carve = [&](size_t bytes) -> void* {
        void* r = (void*)p;
        p += (bytes + 255) & ~(size_t)255;
        return r;
    };
    float*    h1  = (float*)   carve((size_t)N1 * D * 4);
    float*    h2  = (float*)   carve((size_t)N2 * D * 4);
    float*    as1 = (float*)   carve((size_t)N1 * 4);
    float*    ad1 = (float*)   carve((size_t)N1 * 4);
    float*    as2 = (float*)   carve((size_t)N2 * 4);
    float*    ad2 = (float*)   carve((size_t)N2 * 4);
    unsigned* m1  = (unsigned*)carve((size_t)N1 * 4);
    float*    s1  = (float*)   carve((size_t)N1 * 4);
    unsigned* m2  = (unsigned*)carve((size_t)N2 * 4);
    float*    s2  = (float*)   carve((size_t)N2 * 4);
    float*    t1  = (float*)   carve((size_t)N1 * D * 4);
    float*    t2  = (float*)   carve((size_t)N2 * D * 4);

    float* O1 = (float*)d_out;
    float* O2 = O1 + (size_t)N1 * D;

    for (int l = 0; l < L; ++l) {
        const float* in1 = (l == 0) ? X1 : t1;
        const float* in2 = (l == 0) ? X2 : t2;
        float* o1 = (l == L - 1) ? O1 : t1;   // agg accumulates in place, bias added at combine
        float* o2 = (l == L - 1) ? O2 : t2;

        // 1) dense GEMMs + attention logits (x only read here, so in==out buffer is safe)
        gat_gemm<<<(N1 + 127) / 128, 256, 0, stream>>>(in1, W1 + (size_t)l * D * D,
                                                       AS1 + l * D, AD1 + l * D,
                                                       h1, as1, ad1, N1);
        gat_gemm<<<(N2 + 127) / 128, 256, 0, stream>>>(in2, W2 + (size_t)l * D * D,
                                                       AS2 + l * D, AD2 + l * D,
                                                       h2, as2, ad2, N2);

        // 2) zero segment state + aggregation target
        init_seg<<<(N1 * D + 255) / 256, 256, 0, stream>>>(o1, m1, s1, N1);
        init_seg<<<(N2 * D + 255) / 256, 256, 0, stream>>>(o2, m2, s2, N2);

        // 3) segment max
        edge_max_k<<<((E1 + N1) + 255) / 256, 256, 0, stream>>>(src1, dst1, as1, ad1, m1, E1, N1);
        edge_max_k<<<((E2 + N2) + 255) / 256, 256, 0, stream>>>(src2, dst2, as2, ad2, m2, E2, N2);

        // 4) segment exp-sum
        edge_sum_k<<<((E1 + N1) + 255) / 256, 256, 0, stream>>>(src1, dst1, as1, ad1, m1, s1, E1, N1);
        edge_sum_k<<<((E2 + N2) + 255) / 256, 256, 0, stream>>>(src2, dst2, as2, ad2, m2, s2, E2, N2);

        // 5) weighted scatter-add of h[src] into agg[dst]
        edge_agg_k<<<(((size_t)(E1 + N1) * 16 + 255) / 256), 256, 0, stream>>>(
            src1, dst1, as1, ad1, m1, s1, h1, o1, E1, N1);
        edge_agg_k<<<(((size_t)(E2 + N2) * 16 + 255) / 256), 256, 0, stream>>>(
            src2, dst2, as2, ad2, m2, s2, h2, o2, E2, N2);

        // 6) finalize x2 (bias), then x1 (bias + cross-graph gather of final x2)
        combine2_k<<<(N2 * D + 255) / 256, 256, 0, stream>>>(o2, B2 + l * D, N2);
        combine1_k<<<(N1 * D + 255) / 256, 256, 0, stream>>>(o1, B1 + l * D, o2, GRP, N1);
    }
}